// AttenApprox_17832704213548
// MI455X (gfx1250) — compile-verified
//
#include <hip/hip_runtime.h>
#include <hip/hip_bf16.h>

typedef __attribute__((ext_vector_type(2))) float v2f;
typedef __attribute__((ext_vector_type(8))) float v8f;

#define BATCH 2
#define SEQ   2048
#define DMODEL 768
#define HEADS 12
#define FDIM  16
#define DHEAD 64

#define AS1 __attribute__((address_space(1)))
#define AS3 __attribute__((address_space(3)))

#if defined(__has_builtin)
#if __has_builtin(__builtin_amdgcn_global_load_async_to_lds_b32) && \
    __has_builtin(__builtin_amdgcn_s_wait_asynccnt)
#define USE_ASYNC_LDS 1
#endif
#endif

static __device__ __forceinline__ v8f wmma_f32(v2f a, v2f b, v8f c) {
    return __builtin_amdgcn_wmma_f32_16x16x4_f32(false, a, false, b, (short)0, c, false, false);
}

// ---------------------------------------------------------------------------
// Block-tiled fp32 WMMA GEMM: C(MxN) = A(MxK,rm) * B(KxN,rm).
// Block = 128x64 C tile, 8 waves, each wave a 32x32 sub-block (4 WMMA tiles).
// K staged in 16-wide slabs; A row-major (stride 20), B transposed (stride 18)
// in LDS so every WMMA operand is a single aligned ds_load_b64, bank-conflict
// free for both lane halves. Slabs staged via async global->LDS DMA when the
// toolchain exposes it. Requires M%128==0, N%64==0, K%16==0.
// ---------------------------------------------------------------------------
#define GBM 128
#define GBN 64
#define GKT 16

__global__ __launch_bounds__(256) void gemm_wmma_f32(
    const float* __restrict__ A, const float* __restrict__ B,
    float* __restrict__ C, int M, int K, int N)
{
    __shared__ float As[GBM * 20];   // [row][k]  128 x 16, stride 20
    __shared__ float Bt[GBN * 18];   // [col][k]   64 x 16, stride 18 (transposed)

    int tid  = threadIdx.x;
    int wave = tid >> 5, lane = tid & 31;
    int ln15 = lane & 15;
    int dK   = (lane < 16) ? 0 : 2;

    int nBlocks = N / GBN;
    int bm = blockIdx.x / nBlocks;
    int bn = blockIdx.x - bm * nBlocks;
    int rowBase = bm * GBM;
    int colBase = bn * GBN;

    int aRow0 = (wave & 3) * 32;
    int bCol0 = (wave >> 2) * 32;

    int ar = tid >> 4, af = tid & 15;     // A staging: 8 rows apart by 16
    int br = tid >> 6, bc = tid & 63;     // B staging: 4 rows apart by 4

    v8f acc00 = {}, acc01 = {}, acc10 = {}, acc11 = {};

    for (int k0 = 0; k0 < K; k0 += GKT) {
        __syncthreads();
#ifdef USE_ASYNC_LDS
        #pragma unroll
        for (int i = 0; i < 8; ++i)
            __builtin_amdgcn_global_load_async_to_lds_b32(
                (AS1 int*)(A + (size_t)(rowBase + ar + i * 16) * K + k0 + af),
                (AS3 int*)(As + (ar + i * 16) * 20 + af), 0, 0);
        #pragma unroll
        for (int i = 0; i < 4; ++i)
            __builtin_amdgcn_global_load_async_to_lds_b32(
                (AS1 int*)(B + (size_t)(k0 + br + i * 4) * N + colBase + bc),
                (AS3 int*)(Bt + bc * 18 + br + i * 4), 0, 0);
        __builtin_amdgcn_s_wait_asynccnt(0);
#else
        float ta[8], tb[4];
        #pragma unroll
        for (int i = 0; i < 8; ++i)
            ta[i] = A[(size_t)(rowBase + ar + i * 16) * K + k0 + af];
        #pragma unroll
        for (int i = 0; i < 4; ++i)
            tb[i] = B[(size_t)(k0 + br + i * 4) * N + colBase + bc];
        #pragma unroll
        for (int i = 0; i < 8; ++i) As[(ar + i * 16) * 20 + af] = ta[i];
        #pragma unroll
        for (int i = 0; i < 4; ++i) Bt[bc * 18 + br + i * 4] = tb[i];
#endif
        if (k0 + GKT < K) {                                    // global_prefetch_b8
            __builtin_prefetch(&A[(size_t)(rowBase + (tid & 127)) * K + k0 + GKT], 0, 3);
            if (tid < GKT)
                __builtin_prefetch(&B[(size_t)(k0 + GKT + tid) * N + colBase], 0, 3);
        }
        __syncthreads();

        #pragma unroll
        for (int st = 0; st < GKT / 4; ++st) {
            int kk = st * 4 + dK;                              // even -> 8B-aligned
            v2f a0 = *(const v2f*)(&As[(aRow0 + ln15) * 20 + kk]);
            v2f a1 = *(const v2f*)(&As[(aRow0 + 16 + ln15) * 20 + kk]);
            v2f b0 = *(const v2f*)(&Bt[(bCol0 + ln15) * 18 + kk]);
            v2f b1 = *(const v2f*)(&Bt[(bCol0 + 16 + ln15) * 18 + kk]);
            acc00 = wmma_f32(a0, b0, acc00);
            acc01 = wmma_f32(a0, b1, acc01);
            acc10 = wmma_f32(a1, b0, acc10);
            acc11 = wmma_f32(a1, b1, acc11);
        }
    }

    #pragma unroll
    for (int j = 0; j < 8; ++j) {
        int rL = j + ((lane < 16) ? 0 : 8);
        int r0 = rowBase + aRow0 + rL;
        int r1 = r0 + 16;
        int c0 = colBase + bCol0 + ln15;
        int c1 = c0 + 16;
        C[(size_t)r0 * N + c0] = acc00[j];
        C[(size_t)r0 * N + c1] = acc01[j];
        C[(size_t)r1 * N + c0] = acc10[j];
        C[(size_t)r1 * N + c1] = acc11[j];
    }
}

// ---------------------------------------------------------------------------
// Running-mean centering scan. One block per (b,h); thread t<16 scans one k
// feature column, 16<=t<80 scans one v column.
// ---------------------------------------------------------------------------
__global__ __launch_bounds__(128) void center_scan_kernel(
    const float* __restrict__ k_raw, const float* __restrict__ v_raw,
    float* __restrict__ ck, float* __restrict__ cv, float* __restrict__ mv)
{
    int h = blockIdx.x, b = blockIdx.y;
    int t = threadIdx.x;
    int bh = b * HEADS + h;
    if (t < FDIM) {
        const float* src = k_raw + (size_t)b * SEQ * (HEADS * FDIM) + h * FDIM + t;
        float* dst = ck + (size_t)bh * SEQ * FDIM + t;
        float sum = 0.f;
        #pragma unroll 8
        for (int l = 0; l < SEQ; ++l) {
            float x = src[(size_t)l * (HEADS * FDIM)];
            sum += x;
            dst[(size_t)l * FDIM] = x - sum / (float)(l + 1);
        }
    } else if (t < FDIM + DHEAD) {
        int i = t - FDIM;
        const float* src = v_raw + (size_t)b * SEQ * (HEADS * DHEAD) + h * DHEAD + i;
        float* dcv = cv + (size_t)bh * SEQ * DHEAD + i;
        float* dmv = mv + (size_t)bh * SEQ * DHEAD + i;
        float sum = 0.f;
        #pragma unroll 8
        for (int l = 0; l < SEQ; ++l) {
            float x = src[(size_t)l * (HEADS * DHEAD)];
            sum += x;
            float m = sum / (float)(l + 1);
            dmv[(size_t)l * DHEAD] = m;
            dcv[(size_t)l * DHEAD] = x - m;
        }
    }
}

// ---------------------------------------------------------------------------
// Fused causal attention core (flash-style): one block per (b,h,64-query tile).
// S = Q * CK^T via WMMA into LDS (diag tile masked); per-row sum of S^2 and
// O += S * CV (WMMA, CV transposed in LDS) consumed from LDS.
// ---------------------------------------------------------------------------
__global__ __launch_bounds__(256) void atten_kernel(
    const float* __restrict__ q_raw, const float* __restrict__ ck,
    const float* __restrict__ cv, float* __restrict__ qKV, float* __restrict__ rs)
{
    __shared__ float Qs[64 * 20];    // [q][f]   stride 20
    __shared__ float CKs[64 * 20];   // [k][f]   stride 20
    __shared__ float CVt[64 * 68];   // [d][k]   stride 68 (transposed)
    __shared__ float Ss[64 * 68];    // [q][k]   stride 68 (even -> aligned v2f)

    int qt = blockIdx.x;
    int h  = blockIdx.y;
    int b  = blockIdx.z;
    int bh = b * HEADS + h;
    int qBase = qt * 64;

    int tid  = threadIdx.x;
    int wave = tid >> 5, lane = tid & 31;
    int ln15 = lane & 15;
    int dK   = (lane < 16) ? 0 : 2;
    int mt   = wave & 3;
    int ntB  = (wave >> 2) * 2;

    int qr = tid >> 4, qf = tid & 15;     // Q/CK staging: 4 rows apart by 16
    int vr = tid >> 6, vc = tid & 63;     // CV staging: 16 rows apart by 4

    const float* qg = q_raw + ((size_t)(b * SEQ + qBase)) * (HEADS * FDIM) + h * FDIM;
#ifdef USE_ASYNC_LDS
    #pragma unroll
    for (int i = 0; i < 4; ++i)
        __builtin_amdgcn_global_load_async_to_lds_b32(
            (AS1 int*)(qg + (size_t)(qr + i * 16) * (HEADS * FDIM) + qf),
            (AS3 int*)(Qs + (qr + i * 16) * 20 + qf), 0, 0);
#else
    {
        float tq[4];
        #pragma unroll
        for (int i = 0; i < 4; ++i)
            tq[i] = qg[(size_t)(qr + i * 16) * (HEADS * FDIM) + qf];
        #pragma unroll
        for (int i = 0; i < 4; ++i) Qs[(qr + i * 16) * 20 + qf] = tq[i];
    }
#endif

    v8f o0 = {}, o1 = {};
    float rsum = 0.f;

    for (int kt = 0; kt <= qt; ++kt) {
        int kBase = kt * 64;
        const float* ckg = ck + ((size_t)bh * SEQ + kBase) * FDIM;
        const float* cvg = cv + ((size_t)bh * SEQ + kBase) * DHEAD;
        __syncthreads();
#ifdef USE_ASYNC_LDS
        #pragma unroll
        for (int i = 0; i < 4; ++i)
            __builtin_amdgcn_global_load_async_to_lds_b32(
                (AS1 int*)(ckg + (size_t)(qr + i * 16) * FDIM + qf),
                (AS3 int*)(CKs + (qr + i * 16) * 20 + qf), 0, 0);
        #pragma unroll
        for (int i = 0; i < 16; ++i)
            __builtin_amdgcn_global_load_async_to_lds_b32(
                (AS1 int*)(cvg + (size_t)(vr + i * 4) * DHEAD + vc),
                (AS3 int*)(CVt + vc * 68 + vr + i * 4), 0, 0);
        __builtin_amdgcn_s_wait_asynccnt(0);
#else
        {
            float tk[4], tv[16];
            #pragma unroll
            for (int i = 0; i < 4; ++i)
                tk[i] = ckg[(size_t)(qr + i * 16) * FDIM + qf];
            #pragma unroll
            for (int i = 0; i < 16; ++i)
                tv[i] = cvg[(size_t)(vr + i * 4) * DHEAD + vc];
            #pragma unroll
            for (int i = 0; i < 4; ++i) CKs[(qr + i * 16) * 20 + qf] = tk[i];
            #pragma unroll
            for (int i = 0; i < 16; ++i) CVt[vc * 68 + vr + i * 4] = tv[i];
        }
#endif
        __syncthreads();

        // ---- S tiles: two 16x16 tiles per wave, F=16 in four x4 steps ----
        bool diag = (kt == qt);
        #pragma unroll
        for (int sub = 0; sub < 2; ++sub) {
            int nt = ntB + sub;
            v8f s = {};
            #pragma unroll
            for (int st = 0; st < 4; ++st) {
                int f0 = st * 4 + dK;
                v2f a  = *(const v2f*)(&Qs[(mt * 16 + ln15) * 20 + f0]);
                v2f bb = *(const v2f*)(&CKs[(nt * 16 + ln15) * 20 + f0]);
                s = wmma_f32(a, bb, s);
            }
            int c = nt * 16 + ln15;
            #pragma unroll
            for (int j = 0; j < 8; ++j) {
                int m = mt * 16 + j + ((lane < 16) ? 0 : 8);
                float val = s[j];
                if (diag && (c > m)) val = 0.f;   // causal: keep key<=query
                Ss[m * 68 + c] = val;
            }
        }
        __syncthreads();

        // ---- per-row sum of squares (threads 0..63, one row each) ----
        if (tid < 64) {
            float a = 0.f;
            #pragma unroll 8
            for (int c = 0; c < 64; ++c) { float x = Ss[tid * 68 + c]; a += x * x; }
            rsum += a;
        }

        // ---- O += S * CV : contraction over 64 keys, 16 x4 steps ----
        #pragma unroll 4
        for (int st = 0; st < 16; ++st) {
            int k0 = st * 4 + dK;
            v2f a  = *(const v2f*)(&Ss[(mt * 16 + ln15) * 68 + k0]);
            v2f b0 = *(const v2f*)(&CVt[(ntB * 16 + ln15) * 68 + k0]);
            v2f b1 = *(const v2f*)(&CVt[((ntB + 1) * 16 + ln15) * 68 + k0]);
            o0 = wmma_f32(a, b0, o0);
            o1 = wmma_f32(a, b1, o1);
        }
    }

    #pragma unroll
    for (int j = 0; j < 8; ++j) {
        int m = mt * 16 + j + ((lane < 16) ? 0 : 8);
        size_t rowOff = ((size_t)bh * SEQ + qBase + m) * DHEAD;
        qKV[rowOff + ntB * 16 + ln15]       = o0[j];
        qKV[rowOff + (ntB + 1) * 16 + ln15] = o1[j];
    }
    if (tid < 64) rs[(size_t)bh * SEQ + qBase + tid] = rsum;
}

// ---------------------------------------------------------------------------
// Denominator scan: cum = cumsum_q(rowSumSq / (2*DH)); dinv = 1/((q+1)+cum)
// ---------------------------------------------------------------------------
__global__ void denom_kernel(const float* __restrict__ rs, float* __restrict__ dinv)
{
    int bh = blockIdx.x;
    if (threadIdx.x != 0) return;
    const float inv2d = 1.0f / (2.0f * DHEAD);
    float cum = 0.f;
    for (int q = 0; q < SEQ; ++q) {
        cum += rs[(size_t)bh * SEQ + q] * inv2d;
        dinv[(size_t)bh * SEQ + q] = 1.0f / ((float)(q + 1) + cum);
    }
}

// ---------------------------------------------------------------------------
// y[b,l,h*DH+i] = mv + (qKV / sqrt(DH)) * dinv   (row-major for final GEMM)
// ---------------------------------------------------------------------------
__global__ __launch_bounds__(256) void finalize_kernel(
    const float* __restrict__ qKV, const float* __restrict__ mv,
    const float* __restrict__ dinv, float* __restrict__ y)
{
    size_t idx = (size_t)blockIdx.x * 256 + threadIdx.x;
    if (idx >= (size_t)BATCH * HEADS * SEQ * DHEAD) return;
    size_t rowIdx = idx >> 6;           // bh*SEQ + l
    int i  = (int)(idx & 63);
    int l  = (int)(rowIdx & (SEQ - 1));
    int bh = (int)(rowIdx >> 11);
    int b  = bh / HEADS, h = bh - b * HEADS;
    const float invSqrtDh = 0.125f;     // 1/sqrt(64)
    float val = mv[idx] + qKV[idx] * invSqrtDh * dinv[rowIdx];
    y[((size_t)(b * SEQ + l)) * DMODEL + h * DHEAD + i] = val;
}

// ---------------------------------------------------------------------------
extern "C" void kernel_launch(void* const* d_in, const int* in_sizes, int n_in,
                              void* d_out, int out_size, void* d_ws, size_t ws_size,
                              hipStream_t stream) {
    const float* hs = (const float*)d_in[0];
    const float* Wq = (const float*)d_in[1];
    const float* Wk = (const float*)d_in[2];
    const float* Wv = (const float*)d_in[3];
    const float* Wo = (const float*)d_in[4];
    float* out = (float*)d_out;

    const int M  = BATCH * SEQ;            // 4096
    const int NF = HEADS * FDIM;           // 192
    const int NV = HEADS * DHEAD;          // 768

    float* ws = (float*)d_ws;
    size_t off = 0;
    float* q_raw = ws + off; off += (size_t)M * NF;
    float* k_raw = ws + off; off += (size_t)M * NF;
    float* v_raw = ws + off; off += (size_t)M * NV;
    float* ck    = ws + off; off += (size_t)BATCH * HEADS * SEQ * FDIM;
    float* cv    = ws + off; off += (size_t)BATCH * HEADS * SEQ * DHEAD;
    float* mv    = ws + off; off += (size_t)BATCH * HEADS * SEQ * DHEAD;
    float* qKV   = ws + off; off += (size_t)BATCH * HEADS * SEQ * DHEAD;
    float* rsbuf = ws + off; off += (size_t)BATCH * HEADS * SEQ;
    float* dinv  = ws + off; off += (size_t)BATCH * HEADS * SEQ;
    float* ybuf  = ws + off; off += (size_t)M * DMODEL;
    (void)ws_size; (void)in_sizes; (void)n_in; (void)out_size;

    // 1-3: input projections
    gemm_wmma_f32<<<dim3((M / GBM) * (NF / GBN)), 256, 0, stream>>>(hs, Wq, q_raw, M, DMODEL, NF);
    gemm_wmma_f32<<<dim3((M / GBM) * (NF / GBN)), 256, 0, stream>>>(hs, Wk, k_raw, M, DMODEL, NF);
    gemm_wmma_f32<<<dim3((M / GBM) * (NV / GBN)), 256, 0, stream>>>(hs, Wv, v_raw, M, DMODEL, NV);

    // 4: running-mean centering
    center_scan_kernel<<<dim3(HEADS, BATCH), 128, 0, stream>>>(k_raw, v_raw, ck, cv, mv);

    // 5: fused causal attention core
    atten_kernel<<<dim3(SEQ / 64, HEADS, BATCH), 256, 0, stream>>>(q_raw, ck, cv, qKV, rsbuf);

    // 6: denominator scan
    denom_kernel<<<dim3(BATCH * HEADS), 32, 0, stream>>>(rsbuf, dinv);

    // 7: finalize y
    size_t nElem = (size_t)BATCH * HEADS * SEQ * DHEAD;
    finalize_kernel<<<dim3((unsigned)((nElem + 255) / 256)), 256, 0, stream>>>(qKV, mv, dinv, ybuf);

    // 8: output projection
    gemm_wmma_f32<<<dim3((M / GBM) * (DMODEL / GBN)), 256, 0, stream>>>(ybuf, Wo, out, M, DMODEL, DMODEL);
}